// wavesolve_7971459301425
// MI455X (gfx1250) — compile-verified
//
#include <hip/hip_runtime.h>

// 2D wave equation: 512x512 grid, 512 time steps, full trajectory stored.
// Memory-bound (537 MB mandatory HBM writes, ~0.94 GFLOP total -> 1.75 F/B),
// so the design is a single persistent kernel (no 510 kernel launches), with
// per-WG LDS-resident slabs, CDNA5 async LDS<->global movers for halo
// exchange + output streaming, and a global atomic phase barrier per step.

#define WNX 512
#define WNT 512
#define WN2 (WNX * WNX)
#define NWG 128                 // 512 rows / 4 rows per WG
#define WROWS 4
#define WTHREADS 256            // 8 wave32 per WG
#define SLABR (WROWS + 2)       // owned rows + 2 halo rows
#define SLABSZ (SLABR * WNX)

#define ASG __attribute__((address_space(1)))   // global
#define ASL __attribute__((address_space(3)))   // LDS

typedef int wv4i __attribute__((vector_size(16)));   // matches builtin param

#if defined(__HIP_DEVICE_COMPILE__) &&                                   \
    __has_builtin(__builtin_amdgcn_global_load_async_to_lds_b128) &&     \
    __has_builtin(__builtin_amdgcn_global_store_async_from_lds_b128)
#define HAVE_ASYNC_LDS 1
#else
#define HAVE_ASYNC_LDS 0
#endif

__device__ __forceinline__ void wave_wait_async0() {
#if HAVE_ASYNC_LDS
#if __has_builtin(__builtin_amdgcn_s_wait_asynccnt)
  __builtin_amdgcn_s_wait_asynccnt(0);
#else
  asm volatile("s_wait_asynccnt 0" ::: "memory");
#endif
#endif
}

// global -> LDS, 16 bytes, async (ASYNCcnt)
__device__ __forceinline__ void wave_g2l_b128(float* lds_dst, const float* gsrc) {
#if HAVE_ASYNC_LDS
  __builtin_amdgcn_global_load_async_to_lds_b128(
      (ASG wv4i*)(ASG float*)gsrc,
      (ASL wv4i*)(ASL float*)lds_dst, 0, 0);
#else
  *(float4*)lds_dst = *(const float4*)gsrc;
#endif
}

// LDS -> global, 16 bytes, async (ASYNCcnt)
__device__ __forceinline__ void wave_l2g_b128(float* gdst, const float* lds_src) {
#if HAVE_ASYNC_LDS
  __builtin_amdgcn_global_store_async_from_lds_b128(
      (ASG wv4i*)(ASG float*)gdst,
      (ASL wv4i*)(ASL float*)lds_src, 0, 0);
#else
  *(float4*)gdst = *(const float4*)lds_src;
#endif
}

__global__ void wave_ws_init_kernel(unsigned* cnt) {
  const int i = threadIdx.x + blockIdx.x * blockDim.x;
  if (i < WNT) cnt[i] = 0u;
}

__launch_bounds__(WTHREADS)
__global__ void wave_persist_kernel(const float* __restrict__ u0,
                                    const float* __restrict__ alphap,
                                    float* __restrict__ out,
                                    unsigned* __restrict__ cnt) {
  __shared__ __align__(16) float lds[3][SLABSZ];   // 36 KB of 320 KB/WGP
  const int tid = threadIdx.x;
  const int r0  = blockIdx.x * WROWS;

  const float a      = alphap[0];
  const float DXc    = 6.0f / 511.0f;
  const float DTc    = 10.0f / 511.0f;
  const float gamma  = a * a * (DTc * DTc) / (DXc * DXc);
  const float hgamma = 0.5f * gamma;

  int prv = 0, cur = 1, nxt = 2;

  // stream one slab (owned rows, already in LDS) to out[t]
  auto emit_slab = [&](int s, int t) {
    for (int g = tid; g < WROWS * (WNX / 4); g += WTHREADS) {
      const int lr = g >> 7;                 // local owned row
      const int c  = (g & 127) << 2;         // column (x4 floats)
      const int gr = r0 + lr;
      wave_l2g_b128(&out[(size_t)t * WN2 + (size_t)gr * WNX + c],
                    &lds[s][(lr + 1) * WNX + c]);
    }
    wave_wait_async0();
  };

  // pull the two halo rows of slab s from out[t] (published by neighbor WGs)
  auto load_halo = [&](int s, int t) {
    const int half = tid >> 7;               // 0: top halo, 1: bottom halo
    const int c    = (tid & 127) << 2;
    const int gr   = half ? (r0 + WROWS) : (r0 - 1);
    const int lr   = half ? (WROWS + 1) : 0;
    float* dst = &lds[s][lr * WNX + c];
    if (gr >= 0 && gr < WNX) {
      wave_g2l_b128(dst, &out[(size_t)t * WN2 + (size_t)gr * WNX + c]);
    } else {
      *(float4*)dst = make_float4(0.f, 0.f, 0.f, 0.f);
    }
    wave_wait_async0();
    __syncthreads();
  };

  // device-wide phase barrier for step t (counters pre-zeroed each call)
  auto gbar = [&](int t) {
    __threadfence();
    __syncthreads();
    if (tid == 0) {
      __hip_atomic_fetch_add(&cnt[t], 1u, __ATOMIC_ACQ_REL,
                             __HIP_MEMORY_SCOPE_AGENT);
      while (__hip_atomic_load(&cnt[t], __ATOMIC_ACQUIRE,
                               __HIP_MEMORY_SCOPE_AGENT) < (unsigned)gridDim.x) {
        __builtin_amdgcn_s_sleep(1);
      }
    }
    __syncthreads();
  };

  // ---- setup: async-load RAW u0 slab (rows r0-1 .. r0+WROWS) into lds[cur]
  for (int g = tid; g < SLABR * (WNX / 4); g += WTHREADS) {
    const int lr = g >> 7;
    const int c  = (g & 127) << 2;
    const int gr = r0 + lr - 1;
    float* dst = &lds[cur][lr * WNX + c];
    if (gr >= 0 && gr < WNX) wave_g2l_b128(dst, &u0[(size_t)gr * WNX + c]);
    else                     *(float4*)dst = make_float4(0.f, 0.f, 0.f, 0.f);
  }
  wave_wait_async0();
  __syncthreads();

  // ---- step 0: prv = pad0(u0); emit out[0]
  for (int g = tid; g < WROWS * (WNX / 4); g += WTHREADS) {
    const int lr = g >> 7;
    const int c  = (g & 127) << 2;
    const int gr = r0 + lr;
    float4 v = make_float4(0.f, 0.f, 0.f, 0.f);
    if (gr > 0 && gr < WNX - 1) {
      v = *(const float4*)&lds[cur][(lr + 1) * WNX + c];
      if (c == 0)       v.x = 0.f;
      if (c == WNX - 4) v.w = 0.f;
    }
    *(float4*)&lds[prv][(lr + 1) * WNX + c] = v;
  }
  __syncthreads();
  emit_slab(prv, 0);

  // ---- step 1: nxt = pad0(u0_int + 0.5*gamma*lap(RAW u0)); emit out[1]
  for (int g = tid; g < WROWS * (WNX / 4); g += WTHREADS) {
    const int lr = g >> 7;
    const int c  = (g & 127) << 2;
    const int gr = r0 + lr;
    float4 v = make_float4(0.f, 0.f, 0.f, 0.f);
    if (gr > 0 && gr < WNX - 1) {
      float tmp[4];
#pragma unroll
      for (int k = 0; k < 4; ++k) {
        const int cc = c + k;
        if (cc == 0 || cc == WNX - 1) { tmp[k] = 0.f; continue; }
        const float ucv = lds[cur][(lr + 1) * WNX + cc];
        const float lap = lds[cur][lr * WNX + cc] + lds[cur][(lr + 2) * WNX + cc] +
                          lds[cur][(lr + 1) * WNX + cc - 1] +
                          lds[cur][(lr + 1) * WNX + cc + 1] - 4.0f * ucv;
        tmp[k] = ucv + hgamma * lap;
      }
      v = make_float4(tmp[0], tmp[1], tmp[2], tmp[3]);
    }
    *(float4*)&lds[nxt][(lr + 1) * WNX + c] = v;
  }
  __syncthreads();
  emit_slab(nxt, 1);
  gbar(1);
  { const int t0 = cur; cur = nxt; nxt = t0; }  // prv=step0, cur=u1, nxt=recycled

  // ---- steady steps t = 2 .. 511
  for (int t = 2; t < WNT; ++t) {
    load_halo(cur, t - 1);   // halo rows of just-published field
    for (int g = tid; g < WROWS * (WNX / 4); g += WTHREADS) {
      const int lr = g >> 7;
      const int c  = (g & 127) << 2;
      const int gr = r0 + lr;
      float4 v = make_float4(0.f, 0.f, 0.f, 0.f);
      if (gr > 0 && gr < WNX - 1) {
        float tmp[4];
#pragma unroll
        for (int k = 0; k < 4; ++k) {
          const int cc = c + k;
          if (cc == 0 || cc == WNX - 1) { tmp[k] = 0.f; continue; }
          const float ucv = lds[cur][(lr + 1) * WNX + cc];
          const float lap = lds[cur][lr * WNX + cc] + lds[cur][(lr + 2) * WNX + cc] +
                            lds[cur][(lr + 1) * WNX + cc - 1] +
                            lds[cur][(lr + 1) * WNX + cc + 1] - 4.0f * ucv;
          tmp[k] = gamma * lap + 2.0f * ucv - lds[prv][(lr + 1) * WNX + cc];
        }
        v = make_float4(tmp[0], tmp[1], tmp[2], tmp[3]);
      }
      *(float4*)&lds[nxt][(lr + 1) * WNX + c] = v;
    }
    __syncthreads();
    emit_slab(nxt, t);
    if (t < WNT - 1) gbar(t);
    const int t0 = prv; prv = cur; cur = nxt; nxt = t0;
  }
}

extern "C" void kernel_launch(void* const* d_in, const int* in_sizes, int n_in,
                              void* d_out, int out_size, void* d_ws, size_t ws_size,
                              hipStream_t stream) {
  (void)in_sizes; (void)n_in; (void)out_size; (void)ws_size;
  const float* u0    = (const float*)d_in[0];
  const float* alpha = (const float*)d_in[1];
  float*    out = (float*)d_out;
  unsigned* cnt = (unsigned*)d_ws;  // WNT counters, zeroed every call

  wave_ws_init_kernel<<<1, WNT, 0, stream>>>(cnt);
  wave_persist_kernel<<<NWG, WTHREADS, 0, stream>>>(u0, alpha, out, cnt);
}